// Block_76974403879353
// MI455X (gfx1250) — compile-verified
//
#include <hip/hip_runtime.h>
#include <math.h>

// ---- problem constants (match reference) ----
#define B_    32
#define D_    2048
#define H_    16
#define KVH_  4
#define HD_   128
#define MAXS_ 4096
#define NREP_ (H_ / KVH_)
#define FF_   (4 * D_)
#define EPS_  1e-5f
#define NSATT 4          // attention split-KV factor

typedef __attribute__((ext_vector_type(2))) float v2f;
typedef __attribute__((ext_vector_type(4))) float v4f;
typedef __attribute__((ext_vector_type(8))) float v8f;

// ---------------------------------------------------------------------------
// RMSNorm: out[row] = x[row] * g * rsqrt(mean(x^2)+eps). One block per row.
// ---------------------------------------------------------------------------
__global__ void rmsnorm_kernel(const float* __restrict__ x,
                               const float* __restrict__ g,
                               float* __restrict__ out) {
  __shared__ float red[8];
  const int row = blockIdx.x;
  const float* xr = x + row * D_;
  float ss = 0.f;
  for (int i = threadIdx.x; i < D_; i += 256) { float v = xr[i]; ss += v * v; }
  for (int m = 16; m; m >>= 1) ss += __shfl_xor(ss, m, 32);
  const int wave = threadIdx.x >> 5, lane = threadIdx.x & 31;
  if (lane == 0) red[wave] = ss;
  __syncthreads();
  float tot = 0.f;
  for (int w = 0; w < 8; ++w) tot += red[w];
  const float rs = rsqrtf(tot * (1.0f / (float)D_) + EPS_);
  for (int i = threadIdx.x; i < D_; i += 256)
    out[row * D_ + i] = xr[i] * g[i] * rs;
}

// ---------------------------------------------------------------------------
// C[32,N] = A[32,K] @ W[K,N]. fp32 WMMA 16x16x4, split-K via gridDim.y.
// Block = 128 threads (4 waves). Wave -> 16-col stripe, both 16-row M tiles.
// If gridDim.y > 1: each z-block writes a 32xN partial at dst + z*32*N
// (reduced later); else dst is the final output and `resid` may be fused.
// A-frag (ISA 7.12.2, 32-bit A 16x4): lanes 0-15 K={k,k+1}, lanes 16-31
// K={k+2,k+3}; M = lane&15. C/D: VGPR v -> row v / v+8, N = lane&15.
// ---------------------------------------------------------------------------
__global__ void gemm32_wmma(const float* __restrict__ A,
                            const float* __restrict__ W,
                            const float* __restrict__ resid,
                            float* __restrict__ dst,
                            int N, int K) {
  const int wave = threadIdx.x >> 5;
  const int lane = threadIdx.x & 31;
  const int half = lane >> 4;           // 0: lanes 0-15, 1: lanes 16-31
  const int m16  = lane & 15;
  const int col  = blockIdx.x * 64 + wave * 16 + m16;

  const int ksplit = gridDim.y;
  const int kchunk = K / ksplit;        // all K are multiples of 4*ksplit
  const int kbeg   = blockIdx.y * kchunk;
  const int kend   = kbeg + kchunk;
  float* out = (ksplit > 1) ? dst + (size_t)blockIdx.y * B_ * N : dst;
  const float* res = (ksplit > 1) ? nullptr : resid;

  v8f acc0 = {};                         // rows 0..15
  v8f acc1 = {};                         // rows 16..31

  const float* A0 = A + m16 * K + half * 2;
  const float* A1 = A + (m16 + 16) * K + half * 2;
  const float* Wc = W + (size_t)(half * 2) * N + col;

#pragma unroll 4
  for (int k = kbeg; k < kend; k += 4) {
    v2f a0, a1, b;
    a0.x = A0[k];  a0.y = A0[k + 1];
    a1.x = A1[k];  a1.y = A1[k + 1];
    // streaming weights: non-temporal (single-use, > L2 capacity)
    b.x = __builtin_nontemporal_load(Wc + (size_t)k * N);
    b.y = __builtin_nontemporal_load(Wc + (size_t)(k + 1) * N);
    acc0 = __builtin_amdgcn_wmma_f32_16x16x4_f32(false, a0, false, b,
                                                 (short)0, acc0, false, false);
    acc1 = __builtin_amdgcn_wmma_f32_16x16x4_f32(false, a1, false, b,
                                                 (short)0, acc1, false, false);
  }

  const int rbase = half * 8;
  for (int v = 0; v < 8; ++v) {
    const int r0 = rbase + v;
    float c0 = acc0[v], c1 = acc1[v];
    if (res) {
      c0 += res[(size_t)r0 * N + col];
      c1 += res[(size_t)(r0 + 16) * N + col];
    }
    out[(size_t)r0 * N + col]        = c0;
    out[(size_t)(r0 + 16) * N + col] = c1;
  }
}

// ---------------------------------------------------------------------------
// Split-K reduce: out[i] = (resid ? resid[i] : 0) + sum_z partial[z*32*N + i]
// ---------------------------------------------------------------------------
__global__ void ksplit_reduce(const float* __restrict__ partial,
                              const float* __restrict__ resid,
                              float* __restrict__ out, int N, int nsplit) {
  const int i = blockIdx.x * 256 + threadIdx.x;
  const int n = B_ * N;
  if (i >= n) return;
  float acc = resid ? resid[i] : 0.f;
  for (int z = 0; z < nsplit; ++z) acc += partial[(size_t)z * n + i];
  out[i] = acc;
}

// ---------------------------------------------------------------------------
// RoPE on the fresh K and V rows (reference ropes BOTH k and v, not q).
// xk/xv: [B, KVH, HD]. cos/sin: [HD/2]. grid = B*KVH blocks x 64 threads.
// ---------------------------------------------------------------------------
__global__ void rope_kernel(float* __restrict__ xk, float* __restrict__ xv,
                            const float* __restrict__ cosv,
                            const float* __restrict__ sinv) {
  const int bh = blockIdx.x;
  const int i  = threadIdx.x;            // pair index 0..63
  const float c = cosv[i], s = sinv[i];
  float* kp = xk + bh * HD_ + 2 * i;
  float* vp = xv + bh * HD_ + 2 * i;
  const float k0 = kp[0], k1 = kp[1];
  kp[0] = k0 * c - k1 * s;  kp[1] = k0 * s + k1 * c;
  const float v0 = vp[0], v1 = vp[1];
  vp[0] = v0 * c - v1 * s;  vp[1] = v0 * s + v1 * c;
}

// ---------------------------------------------------------------------------
// Split-KV flash-decode attention. grid = (B*KVH, NSATT). Each block handles
// its key chunk for the 4 query heads sharing one kv head, does online
// softmax per wave + LDS combine, and emits unnormalized partials
// (block max m, scaled sum s, unnormalized y) for the combine kernel.
// New token (j == start_pos) comes from ws (inputs immutable -> no cache copy).
// ---------------------------------------------------------------------------
__global__ void attn_kernel(const float* __restrict__ q,    // [B,H,HD]
                            const float* __restrict__ kc,   // [B,MAXS,KVH,HD]
                            const float* __restrict__ vc,
                            const float* __restrict__ xk,   // [B,KVH,HD] roped
                            const float* __restrict__ xv,
                            float* __restrict__ pm,         // [B*H, NS]
                            float* __restrict__ ps,         // [B*H, NS]
                            float* __restrict__ py,         // [B*H, NS, HD]
                            const int* __restrict__ sp) {
  const int b    = blockIdx.x / KVH_;
  const int kvh  = blockIdx.x % KVH_;
  const int z    = blockIdx.y;
  const int NS   = gridDim.y;
  const int wave = threadIdx.x >> 5;
  const int lane = threadIdx.x & 31;
  const int start_pos = *sp;
  const int klen = start_pos + 1;
  const int chunk = (klen + NS - 1) / NS;
  const int jbeg = z * chunk;
  const int jend = min(klen, jbeg + chunk);
  const float scale = 0.08838834764831845f;   // 128^-0.5

  __shared__ __align__(16) float ypart[8][NREP_][HD_];  // 16 KB
  __shared__ float wm[8][NREP_];
  __shared__ float wsum[8][NREP_];

  // q fragments: 4 heads x 4 dims per lane
  v4f qf[NREP_];
  for (int r = 0; r < NREP_; ++r)
    qf[r] = *(const v4f*)(q + (size_t)((b * H_ + kvh * NREP_ + r) * HD_) + lane * 4);

  float m[NREP_], s[NREP_];
  v4f yacc[NREP_];
  for (int r = 0; r < NREP_; ++r) { m[r] = -3.0e38f; s[r] = 0.f; yacc[r] = (v4f){0.f,0.f,0.f,0.f}; }

  const float* xkp = xk + (size_t)(b * KVH_ + kvh) * HD_ + lane * 4;
  const float* xvp = xv + (size_t)(b * KVH_ + kvh) * HD_ + lane * 4;

  for (int j = jbeg + wave; j < jend; j += 8) {
    const size_t coff = (((size_t)b * MAXS_ + j) * KVH_ + kvh) * HD_ + lane * 4;
    const bool fresh = (j == start_pos);
    const v4f kv = fresh ? *(const v4f*)xkp
                         : __builtin_nontemporal_load((const v4f*)(kc + coff));
    const v4f vv = fresh ? *(const v4f*)xvp
                         : __builtin_nontemporal_load((const v4f*)(vc + coff));
    float d[NREP_];
    for (int r = 0; r < NREP_; ++r)
      d[r] = qf[r].x * kv.x + qf[r].y * kv.y + qf[r].z * kv.z + qf[r].w * kv.w;
    for (int msk = 16; msk; msk >>= 1)
      for (int r = 0; r < NREP_; ++r) d[r] += __shfl_xor(d[r], msk, 32);
    for (int r = 0; r < NREP_; ++r) {
      const float lg = d[r] * scale;
      const float mn = fmaxf(m[r], lg);
      const float corr = __expf(m[r] - mn);
      const float p    = __expf(lg - mn);
      s[r] = s[r] * corr + p;
      yacc[r].x = yacc[r].x * corr + p * vv.x;
      yacc[r].y = yacc[r].y * corr + p * vv.y;
      yacc[r].z = yacc[r].z * corr + p * vv.z;
      yacc[r].w = yacc[r].w * corr + p * vv.w;
      m[r] = mn;
    }
  }

  for (int r = 0; r < NREP_; ++r) {
    if (lane == 0) { wm[wave][r] = m[r]; wsum[wave][r] = s[r]; }
    *(v4f*)&ypart[wave][r][lane * 4] = yacc[r];
  }
  __syncthreads();

  if (threadIdx.x < HD_) {
    const int dim = threadIdx.x;
    for (int r = 0; r < NREP_; ++r) {
      const int g = b * H_ + kvh * NREP_ + r;
      float M = wm[0][r];
      for (int w = 1; w < 8; ++w) M = fmaxf(M, wm[w][r]);
      float tot = 0.f, acc = 0.f;
      for (int w = 0; w < 8; ++w) {
        const float f = __expf(wm[w][r] - M);
        tot += wsum[w][r] * f;
        acc += ypart[w][r][dim] * f;
      }
      py[((size_t)g * NS + z) * HD_ + dim] = acc;       // unnormalized
      if (dim == 0) { pm[g * NS + z] = M; ps[g * NS + z] = tot; }
    }
  }
}

// ---------------------------------------------------------------------------
// Combine split-KV partials: standard log-sum-exp merge. grid = B*H x 128thr.
// ---------------------------------------------------------------------------
__global__ void attn_combine(const float* __restrict__ pm,
                             const float* __restrict__ ps,
                             const float* __restrict__ py,
                             float* __restrict__ y, int NS) {
  const int g = blockIdx.x;
  const int dim = threadIdx.x;
  float M = -3.0e38f;
  for (int z = 0; z < NS; ++z) M = fmaxf(M, pm[g * NS + z]);
  float tot = 0.f, acc = 0.f;
  for (int z = 0; z < NS; ++z) {
    const float f = __expf(pm[g * NS + z] - M);
    tot += ps[g * NS + z] * f;
    acc += py[((size_t)g * NS + z) * HD_ + dim] * f;
  }
  y[(size_t)g * HD_ + dim] = acc / tot;
}

// ---------------------------------------------------------------------------
// g = silu(t1) * t2
// ---------------------------------------------------------------------------
__global__ void silu_mul_kernel(const float* __restrict__ t1,
                                const float* __restrict__ t2,
                                float* __restrict__ g, int n) {
  const int i = blockIdx.x * 256 + threadIdx.x;
  if (i < n) {
    const float a = t1[i];
    g[i] = (a / (1.f + __expf(-a))) * t2[i];
  }
}

// ---------------------------------------------------------------------------
extern "C" void kernel_launch(void* const* d_in, const int* in_sizes, int n_in,
                              void* d_out, int out_size, void* d_ws, size_t ws_size,
                              hipStream_t stream) {
  const float* x    = (const float*)d_in[0];
  const float* cosv = (const float*)d_in[1];
  const float* sinv = (const float*)d_in[2];
  const float* kc   = (const float*)d_in[3];
  const float* vc   = (const float*)d_in[4];
  const float* wra  = (const float*)d_in[5];
  const float* wq   = (const float*)d_in[6];
  const float* wk   = (const float*)d_in[7];
  const float* wv   = (const float*)d_in[8];
  const float* wo   = (const float*)d_in[9];
  const float* wrf  = (const float*)d_in[10];
  const float* w1   = (const float*)d_in[11];
  const float* w2   = (const float*)d_in[12];
  const float* w3   = (const float*)d_in[13];
  const int*   sp   = (const int*)d_in[14];

  // workspace partition (floats): ~7.8 MB total
  float* xa   = (float*)d_ws;
  float* qv   = xa + B_ * D_;
  float* xk   = qv + B_ * D_;
  float* xv   = xk + B_ * (KVH_ * HD_);
  float* yv   = xv + B_ * (KVH_ * HD_);
  float* hv   = yv + B_ * D_;
  float* xf   = hv + B_ * D_;
  float* t1   = xf + B_ * D_;
  float* t2   = t1 + B_ * FF_;
  float* gg   = t2 + B_ * FF_;
  float* part = gg + B_ * FF_;              // 8*32*2048 == 2*32*8192 floats (2 MB)
  float* pm   = part + 8 * B_ * D_;
  float* ps   = pm + B_ * H_ * NSATT;
  float* py   = ps + B_ * H_ * NSATT;       // B*H*NSATT*HD floats (1 MB)

  // 1) pre-attention RMSNorm
  rmsnorm_kernel<<<B_, 256, 0, stream>>>(x, wra, xa);

  // 2) Q/K/V projections (fp32 WMMA, split-K for block-count / bandwidth)
  gemm32_wmma<<<dim3(D_ / 64, 8), 128, 0, stream>>>(xa, wq, nullptr, part, D_, D_);
  ksplit_reduce<<<(B_ * D_ + 255) / 256, 256, 0, stream>>>(part, nullptr, qv, D_, 8);
  gemm32_wmma<<<dim3((KVH_ * HD_) / 64, 8), 128, 0, stream>>>(xa, wk, nullptr, part, KVH_ * HD_, D_);
  ksplit_reduce<<<(B_ * KVH_ * HD_ + 255) / 256, 256, 0, stream>>>(part, nullptr, xk, KVH_ * HD_, 8);
  gemm32_wmma<<<dim3((KVH_ * HD_) / 64, 8), 128, 0, stream>>>(xa, wv, nullptr, part, KVH_ * HD_, D_);
  ksplit_reduce<<<(B_ * KVH_ * HD_ + 255) / 256, 256, 0, stream>>>(part, nullptr, xv, KVH_ * HD_, 8);

  // 3) RoPE on new K and V rows (reference ropes k AND v)
  rope_kernel<<<B_ * KVH_, 64, 0, stream>>>(xk, xv, cosv, sinv);

  // 4) split-KV flash-decode attention + combine
  attn_kernel<<<dim3(B_ * KVH_, NSATT), 256, 0, stream>>>(qv, kc, vc, xk, xv, pm, ps, py, sp);
  attn_combine<<<B_ * H_, HD_, 0, stream>>>(pm, ps, py, yv, NSATT);

  // 5) output projection + residual: h = x + y @ wo
  gemm32_wmma<<<dim3(D_ / 64, 8), 128, 0, stream>>>(yv, wo, nullptr, part, D_, D_);
  ksplit_reduce<<<(B_ * D_ + 255) / 256, 256, 0, stream>>>(part, x, hv, D_, 8);

  // 6) pre-FFN RMSNorm
  rmsnorm_kernel<<<B_, 256, 0, stream>>>(hv, wrf, xf);

  // 7) FFN up/gate projections (already 128 N-blocks; split-K 2 -> 256 blocks)
  gemm32_wmma<<<dim3(FF_ / 64, 2), 128, 0, stream>>>(xf, w1, nullptr, part, FF_, D_);
  ksplit_reduce<<<(B_ * FF_ + 255) / 256, 256, 0, stream>>>(part, nullptr, t1, FF_, 2);
  gemm32_wmma<<<dim3(FF_ / 64, 2), 128, 0, stream>>>(xf, w2, nullptr, part, FF_, D_);
  ksplit_reduce<<<(B_ * FF_ + 255) / 256, 256, 0, stream>>>(part, nullptr, t2, FF_, 2);
  silu_mul_kernel<<<(B_ * FF_ + 255) / 256, 256, 0, stream>>>(t1, t2, gg, B_ * FF_);

  // 8) down projection + residual: out = h + g @ w3 (64 MB weight, split-K 8)
  gemm32_wmma<<<dim3(D_ / 64, 8), 128, 0, stream>>>(gg, w3, nullptr, part, D_, FF_);
  ksplit_reduce<<<(B_ * D_ + 255) / 256, 256, 0, stream>>>(part, hv, (float*)d_out, D_, 8);
}